// ANFIS_85727547228732
// MI455X (gfx1250) — compile-verified
//
#include <hip/hip_runtime.h>
#include <math.h>

typedef __attribute__((ext_vector_type(2))) float v2f;
typedef __attribute__((ext_vector_type(8))) float v8f;

#define N 8000
#define D 9
#define TILES 500            // 8000 / 16
#define WAVES_PER_BLOCK 8    // 256 threads, wave32

// ---------------------------------------------------------------------------
// Kernel A: per-row reduction.
//   s1[i] = sum_j ((x[i,j]-c1)/a1)^2      (so prod exp(-d^2) = exp(-s1))
//   s2[i] = sum_j ((x[i,j]-c2)/a2)^2
//   f1[i] = x[i,:] . w_fc1 + b_fc1
//   f2[i] = x[i,:] . w_fc2 + b_fc2
// ---------------------------------------------------------------------------
__global__ __launch_bounds__(256) void anfis_rowstats(
    const float* __restrict__ x,
    const float* __restrict__ a1p, const float* __restrict__ c1p,
    const float* __restrict__ a2p, const float* __restrict__ c2p,
    const float* __restrict__ wfc1, const float* __restrict__ bfc1,
    const float* __restrict__ wfc2, const float* __restrict__ bfc2,
    float* __restrict__ s1, float* __restrict__ s2,
    float* __restrict__ f1, float* __restrict__ f2)
{
    int i = blockIdx.x * blockDim.x + threadIdx.x;
    if (i >= N) return;

    const float inva1 = 1.0f / a1p[0];
    const float inva2 = 1.0f / a2p[0];
    const float c1 = c1p[0];
    const float c2 = c2p[0];

    float acc1 = 0.0f, acc2 = 0.0f, dot1 = 0.0f, dot2 = 0.0f;
#pragma unroll
    for (int j = 0; j < D; ++j) {
        float v  = x[i * D + j];
        float d1 = (v - c1) * inva1;
        float d2 = (v - c2) * inva2;
        acc1 += d1 * d1;
        acc2 += d2 * d2;
        dot1 += v * wfc1[j];
        dot2 += v * wfc2[j];
    }
    s1[i] = acc1;
    s2[i] = acc2;
    f1[i] = dot1 + bfc1[0];
    f2[i] = dot2 + bfc2[0];
}

// ---------------------------------------------------------------------------
// Kernel B: single-block inclusive scan of (s1,s2) over 8000 rows, then
//   w1 = exp(-cumsum(s1)), w2 = exp(-cumsum(s2)), w{1,2}b = w/(w1+w2).
// Chunked Hillis-Steele scan in LDS with a running carry (8 chunks of 1024).
// ---------------------------------------------------------------------------
__global__ __launch_bounds__(1024) void anfis_scan_weights(
    const float* __restrict__ s1, const float* __restrict__ s2,
    float* __restrict__ w1b, float* __restrict__ w2b)
{
    __shared__ float b1[1024];
    __shared__ float b2[1024];

    const int tid = threadIdx.x;
    float carry1 = 0.0f, carry2 = 0.0f;

    for (int base = 0; base < N; base += 1024) {
        int i = base + tid;
        float v1 = (i < N) ? s1[i] : 0.0f;
        float v2 = (i < N) ? s2[i] : 0.0f;
        b1[tid] = v1;
        b2[tid] = v2;
        __syncthreads();

        for (int off = 1; off < 1024; off <<= 1) {
            float t1 = 0.0f, t2 = 0.0f;
            if (tid >= off) { t1 = b1[tid - off]; t2 = b2[tid - off]; }
            __syncthreads();
            b1[tid] += t1;
            b2[tid] += t2;
            __syncthreads();
        }

        if (i < N) {
            float S1 = carry1 + b1[tid];
            float S2 = carry2 + b2[tid];
            float w1 = __expf(-S1);
            float w2 = __expf(-S2);
            float inv = 1.0f / (w1 + w2);
            w1b[i] = w1 * inv;
            w2b[i] = w2 * inv;
        }
        carry1 += b1[1023];
        carry2 += b2[1023];
        __syncthreads();   // protect LDS before next chunk's writes
    }
}

// ---------------------------------------------------------------------------
// Kernel C: out[i,j] = f1[i]*w1b[j] + f2[i]*w2b[j]  as a rank-2 GEMM tile op
// via V_WMMA_F32_16X16X4_F32 (K padded 2 -> 4 with zeros).
//
// A (16x4) lane layout: lanes 0-15 hold K=0,1 ; lanes 16-31 hold K=2,3.
//   -> put f1 in K=0 (lanes 0-15), f2 in K=2 (lanes 16-31), K=1,3 = 0.
// B (4x16) symmetric: w1b in K=0, w2b in K=2.
// C/D (16x16 f32): VGPR r -> (M=r, N=lane) for lanes 0-15,
//                            (M=r+8, N=lane-16) for lanes 16-31.
// One wave per 16x16 output tile; 8 waves per block; EXEC all-1s throughout.
// ---------------------------------------------------------------------------
__global__ __launch_bounds__(256) void anfis_outer_wmma(
    const float* __restrict__ f1, const float* __restrict__ f2,
    const float* __restrict__ w1b, const float* __restrict__ w2b,
    float* __restrict__ out)
{
    const int wave   = blockIdx.x * WAVES_PER_BLOCK + (threadIdx.x >> 5);
    const int lane   = threadIdx.x & 31;
    const int half   = lane >> 4;       // 0: K=0,1 ; 1: K=2,3
    const int m      = lane & 15;

    const int tile_i = wave / TILES;
    const int tile_j = wave % TILES;
    const int row0   = tile_i * 16;
    const int col0   = tile_j * 16;

    // Branch-free operand selection: both half-waves load valid addresses.
    const float* pa = half ? f2 : f1;   // A: K0 <- f1, K2 <- f2
    const float* pb = half ? w2b : w1b; // B: K0 <- w1b, K2 <- w2b

    v2f a;
    a.x = pa[row0 + m];
    a.y = 0.0f;                          // K=1 / K=3 are zero
    v2f b;
    b.x = pb[col0 + m];
    b.y = 0.0f;

    v8f c = {};
    v8f d = __builtin_amdgcn_wmma_f32_16x16x4_f32(
        /*neg_a=*/false, a, /*neg_b=*/false, b,
        /*c_mod=*/(short)0, c, /*reuse_a=*/false, /*reuse_b=*/false);

    // Store the 16x16 tile: VGPR r covers rows r (lanes 0-15) / r+8 (16-31).
    const int col = col0 + m;
#pragma unroll
    for (int r = 0; r < 8; ++r) {
        int row = row0 + r + 8 * half;
        out[(long)row * N + col] = d[r];
    }
}

// ---------------------------------------------------------------------------
extern "C" void kernel_launch(void* const* d_in, const int* in_sizes, int n_in,
                              void* d_out, int out_size, void* d_ws, size_t ws_size,
                              hipStream_t stream) {
    (void)in_sizes; (void)n_in; (void)out_size; (void)ws_size;

    const float* x    = (const float*)d_in[0];
    const float* a1   = (const float*)d_in[1];
    const float* c1   = (const float*)d_in[2];
    const float* a2   = (const float*)d_in[3];
    const float* c2   = (const float*)d_in[4];
    const float* wfc1 = (const float*)d_in[5];
    const float* bfc1 = (const float*)d_in[6];
    const float* wfc2 = (const float*)d_in[7];
    const float* bfc2 = (const float*)d_in[8];
    float* out = (float*)d_out;

    // Workspace layout: 6 arrays of N floats.
    float* ws  = (float*)d_ws;
    float* s1  = ws + 0 * N;
    float* s2  = ws + 1 * N;
    float* f1  = ws + 2 * N;
    float* f2  = ws + 3 * N;
    float* w1b = ws + 4 * N;
    float* w2b = ws + 5 * N;

    // A: per-row stats (8000 rows, 1 thread/row)
    anfis_rowstats<<<(N + 255) / 256, 256, 0, stream>>>(
        x, a1, c1, a2, c2, wfc1, bfc1, wfc2, bfc2, s1, s2, f1, f2);

    // B: single-block scan + weight normalization
    anfis_scan_weights<<<1, 1024, 0, stream>>>(s1, s2, w1b, w2b);

    // C: 500x500 tiles of 16x16, one wave each, 8 waves per block
    anfis_outer_wmma<<<(TILES * TILES) / WAVES_PER_BLOCK, 256, 0, stream>>>(
        f1, f2, w1b, w2b, out);
}